// Tok_k_8504035246112
// MI455X (gfx1250) — compile-verified
//
#include <hip/hip_runtime.h>
#include <math.h>
#include <stdint.h>

// ---------------------------------------------------------------------------
// Top-k (k=8) row mask: keep the 8 largest values per 4096-wide row, set the
// rest to -inf.  x: [16384, 4096] fp32.
//
// One wave32 per row. Stage the 16 KB row into LDS with CDNA5 async
// global->LDS b128 loads (ASYNCcnt), branchless chunk-of-8 selection
// (Batcher sort-8 + bitonic top-8 merge), wave32 xor-butterfly to get the
// row threshold in-register, then masked write-back with non-temporal
// b128 stores.
// ---------------------------------------------------------------------------

#define COLS 4096
#define TPB  32          // one wave32 per row

typedef float v4f __attribute__((ext_vector_type(4)));

// compare-exchange, descending (hi to lower index)
#define CE(a, b)                                                               \
  do {                                                                         \
    float _hi = fmaxf(a, b);                                                   \
    float _lo = fminf(a, b);                                                   \
    (a) = _hi;                                                                 \
    (b) = _lo;                                                                 \
  } while (0)

// Batcher odd-even mergesort, n=8, 19 comparators, descending
__device__ __forceinline__ void sort8_desc(float (&s)[8]) {
  CE(s[0], s[1]); CE(s[2], s[3]); CE(s[4], s[5]); CE(s[6], s[7]);
  CE(s[0], s[2]); CE(s[1], s[3]); CE(s[4], s[6]); CE(s[5], s[7]);
  CE(s[1], s[2]); CE(s[5], s[6]);
  CE(s[0], s[4]); CE(s[1], s[5]); CE(s[2], s[6]); CE(s[3], s[7]);
  CE(s[2], s[4]); CE(s[3], s[5]);
  CE(s[1], s[2]); CE(s[3], s[4]); CE(s[5], s[6]);
}

// bitonic clean-up of a length-8 bitonic sequence, descending (12 CE)
__device__ __forceinline__ void bitonic8_desc(float (&o)[8]) {
  CE(o[0], o[4]); CE(o[1], o[5]); CE(o[2], o[6]); CE(o[3], o[7]);
  CE(o[0], o[2]); CE(o[1], o[3]); CE(o[4], o[6]); CE(o[5], o[7]);
  CE(o[0], o[1]); CE(o[2], o[3]); CE(o[4], o[5]); CE(o[6], o[7]);
}

// merge sorted-desc s[8] into running sorted-desc top-8 t[8]
__device__ __forceinline__ void merge_top8(float (&t)[8], const float (&s)[8]) {
  float o[8];
#pragma unroll
  for (int i = 0; i < 8; ++i) o[i] = fmaxf(t[i], s[7 - i]);  // bitonic
  bitonic8_desc(o);
#pragma unroll
  for (int i = 0; i < 8; ++i) t[i] = o[i];
}

// xor-butterfly step: merge this lane's sorted top-8 with lane^mask partner's
__device__ __forceinline__ void merge_xor(float (&t)[8], int mask) {
  float o[8];
#pragma unroll
  for (int i = 0; i < 8; ++i) {
    float p = __shfl_xor(t[7 - i], mask, 32);
    o[i] = fmaxf(t[i], p);
  }
  bitonic8_desc(o);
#pragma unroll
  for (int i = 0; i < 8; ++i) t[i] = o[i];
}

#define ALOAD(off)                                                             \
  "global_load_async_to_lds_b128 %0, %1, off offset:" #off "\n\t"

__global__ __launch_bounds__(TPB) void topk8_mask_kernel(
    const float* __restrict__ x, float* __restrict__ out) {
  __shared__ float smem[COLS];  // 16 KB row staging

  const int lane = threadIdx.x;
  const size_t row = blockIdx.x;

  // ---- Stage the row into LDS: 32 async b128 bursts (512 B each) ----------
  // INST_OFFSET is added to BOTH the LDS and the global address.
  unsigned lds_addr =
      (unsigned)(uintptr_t)(&smem[0]) + (unsigned)lane * 16u;
  unsigned long long gaddr =
      (unsigned long long)(uintptr_t)(x + row * COLS) +
      (unsigned long long)lane * 16ull;

  asm volatile(
      "global_load_async_to_lds_b128 %0, %1, off\n\t"
      ALOAD(512)   ALOAD(1024)  ALOAD(1536)  ALOAD(2048)
      ALOAD(2560)  ALOAD(3072)  ALOAD(3584)  ALOAD(4096)
      ALOAD(4608)  ALOAD(5120)  ALOAD(5632)  ALOAD(6144)
      ALOAD(6656)  ALOAD(7168)  ALOAD(7680)  ALOAD(8192)
      ALOAD(8704)  ALOAD(9216)  ALOAD(9728)  ALOAD(10240)
      ALOAD(10752) ALOAD(11264) ALOAD(11776) ALOAD(12288)
      ALOAD(12800) ALOAD(13312) ALOAD(13824) ALOAD(14336)
      ALOAD(14848) ALOAD(15360) ALOAD(15872)
      :
      : "v"(lds_addr), "v"(gaddr)
      : "memory");
  // Same wave consumes its own async loads: wait ASYNCcnt==0, no barrier
  // needed (single-wave workgroup; barrier would be an S_NOP anyway).
  asm volatile("s_wait_asynccnt 0" ::: "memory");

  // ---- Branchless top-8 selection, chunks of 8 ---------------------------
  // Lane L, iteration r reads float4 at element r*128 + 4L (bank-friendly,
  // 512 B contiguous per wave-iteration).
  float t[8];
#pragma unroll
  for (int i = 0; i < 8; ++i) t[i] = -INFINITY;

#pragma unroll
  for (int c = 0; c < 16; ++c) {
    v4f a = *(const v4f*)(&smem[(2 * c + 0) * 128 + lane * 4]);
    v4f b = *(const v4f*)(&smem[(2 * c + 1) * 128 + lane * 4]);
    float s[8] = {a[0], a[1], a[2], a[3], b[0], b[1], b[2], b[3]};
    sort8_desc(s);        // 19 CE
    merge_top8(t, s);     // 8 max + 12 CE
  }

  // ---- wave32 xor-butterfly: whole row reduced within one wave -----------
  merge_xor(t, 1);
  merge_xor(t, 2);
  merge_xor(t, 4);
  merge_xor(t, 8);
  merge_xor(t, 16);

  const float thr = t[7];  // row's 8th-largest value, uniform across lanes

  // ---- Masked write-back: keep v >= threshold, else -inf ------------------
  // (ties at the threshold are vanishingly unlikely for continuous inputs)
  float* __restrict__ orow = out + row * COLS;
#pragma unroll 8
  for (int r = 0; r < 32; ++r) {
    int base = r * 128 + lane * 4;
    v4f v = *(const v4f*)(&smem[base]);  // ds_load_b128
    v4f o;
#pragma unroll
    for (int cpt = 0; cpt < 4; ++cpt)
      o[cpt] = (v[cpt] >= thr) ? v[cpt] : -INFINITY;
    __builtin_nontemporal_store(o, (v4f*)(orow + base));  // streaming b128
  }
}

extern "C" void kernel_launch(void* const* d_in, const int* in_sizes, int n_in,
                              void* d_out, int out_size, void* d_ws,
                              size_t ws_size, hipStream_t stream) {
  (void)n_in; (void)d_ws; (void)ws_size; (void)out_size;
  const float* x = (const float*)d_in[0];
  // d_in[1] is k (==8); kernel is specialized for k=8.
  float* out = (float*)d_out;
  const int rows = in_sizes[0] / COLS;  // 16384
  topk8_mask_kernel<<<rows, TPB, 0, stream>>>(x, out);
}